// FilteredPatchLoss_33775622815855
// MI455X (gfx1250) — compile-verified
//
#include <hip/hip_runtime.h>

// CDNA5 / gfx1250 native vector types (WMMA operands + non-temporal loads).
typedef __attribute__((ext_vector_type(2))) float v2f;  // 16x4 f32 A/B: 64 vals / 32 lanes
typedef __attribute__((ext_vector_type(4))) float v4f;  // b128 load payload
typedef __attribute__((ext_vector_type(8))) float v8f;  // 16x16 f32 C/D

namespace {
constexpr int kW          = 1024;
constexpr int kP          = 16;
constexpr int kStripVec4  = (kP * kW) / 4;      // 4096 float4 per 16-row strip
constexpr int kBlocks     = 64 * (1024 / kP);   // 64 batches * 64 patch-rows = 4096 strips
}

// Stage 1: each block reduces one patch-row strip (16 rows x 1024 cols) of one
// batch image = 64 patches. Fully coalesced non-temporal b128 loads (data is
// streamed exactly once -> don't pollute L2/WGP$); per-patch reduction via
// V_WMMA_F32_16X16X4_F32 (A = 16 patches x 4 partials, B = ones -> row sums).
__global__ __launch_bounds__(256)
void fpl_stage1(const float* __restrict__ outp,
                const float* __restrict__ tgtp,
                const int*   __restrict__ filter_rate,
                float*       __restrict__ partials)
{
    __shared__ float s_t[256];   // per-thread target partial sums
    __shared__ float s_l[256];   // per-thread |o-t| partial sums

    const int t   = threadIdx.x;
    const int blk = blockIdx.x;

    const v4f* o4 = reinterpret_cast<const v4f*>(outp) + (size_t)blk * kStripVec4 + t;
    const v4f* t4 = reinterpret_cast<const v4f*>(tgtp) + (size_t)blk * kStripVec4 + t;

    // Thread t owns float4-column t for all 16 rows of the strip; this column
    // lies entirely inside patch t/4. Each iteration: block reads 4KB contiguous per array.
    float st = 0.f, sl = 0.f;
#pragma unroll
    for (int r = 0; r < 16; ++r) {
        const v4f a = __builtin_nontemporal_load(&o4[r * 256]);
        const v4f b = __builtin_nontemporal_load(&t4[r * 256]);
        st += (b.x + b.y) + (b.z + b.w);
        sl += (fabsf(a.x - b.x) + fabsf(a.y - b.y)) +
              (fabsf(a.z - b.z) + fabsf(a.w - b.w));
    }
    s_t[t] = st;
    s_l[t] = sl;
    __syncthreads();

    // Wave 0 (threads 0..31, EXEC all ones as required by WMMA) reduces the
    // 256 partials -> 64 patch sums using the matrix unit, then masks+counts.
    if (t < 32) {
        const float thresh = 256.0f * (float)(*filter_rate);  // t_mean>fr <=> t_sum>256*fr
        const int m  = t & 15;   // A-matrix row (patch within group)
        const int kh = t >> 4;   // lanes 0-15 hold K=0,1 ; lanes 16-31 hold K=2,3
        const v2f ones = {1.0f, 1.0f};
        float loss = 0.f, cnt = 0.f;

#pragma unroll
        for (int pg = 0; pg < 4; ++pg) {          // 4 groups of 16 patches
            const int idx = (pg * 16 + m) * 4 + 2 * kh;
            v2f at = { s_t[idx], s_t[idx + 1] };
            v2f al = { s_l[idx], s_l[idx + 1] };
            v8f ct = {};
            v8f cl = {};
            // D[m][n] = sum_k A[m][k] : per-patch sums replicated across N.
            ct = __builtin_amdgcn_wmma_f32_16x16x4_f32(false, at, false, ones,
                                                       (short)0, ct, false, false);
            cl = __builtin_amdgcn_wmma_f32_16x16x4_f32(false, al, false, ones,
                                                       (short)0, cl, false, false);
            // D layout: VGPR r, lanes 0-15 -> M=r ; lanes 16-31 -> M=r+8.
            // So lane 0 holds patches pg*16+0..7, lane 16 holds pg*16+8..15.
            if ((t & 15) == 0) {
#pragma unroll
                for (int i = 0; i < 8; ++i) {
                    if (ct[i] > thresh) { loss += cl[i]; cnt += 1.0f; }
                }
            }
        }
        // Combine lane 0 + lane 16 contributions (wave32).
        loss += __shfl_xor(loss, 16, 32);
        cnt  += __shfl_xor(cnt, 16, 32);
        if (t == 0) {
            partials[2 * blk]     = loss;  // raw |o-t| sum over masked patches
            partials[2 * blk + 1] = cnt;   // masked patch count
        }
    }
}

// Stage 2: deterministic tree reduction of 4096 (loss,count) pairs -> scalar.
__global__ __launch_bounds__(256)
void fpl_stage2(const float* __restrict__ partials, float* __restrict__ out)
{
    __shared__ float rl[256];
    __shared__ float rc[256];
    const int t = threadIdx.x;
    float l = 0.f, c = 0.f;
    for (int i = t; i < kBlocks; i += 256) {
        l += partials[2 * i];
        c += partials[2 * i + 1];
    }
    rl[t] = l;
    rc[t] = c;
    __syncthreads();
    for (int s = 128; s > 0; s >>= 1) {
        if (t < s) { rl[t] += rl[t + s]; rc[t] += rc[t + s]; }
        __syncthreads();
    }
    if (t == 0) {
        // loss_sum (of patch means) = raw_sum / 256 ; result = loss_sum / count
        out[0] = (rl[0] * (1.0f / 256.0f)) / rc[0];
    }
}

extern "C" void kernel_launch(void* const* d_in, const int* in_sizes, int n_in,
                              void* d_out, int out_size, void* d_ws, size_t ws_size,
                              hipStream_t stream)
{
    const float* outp  = (const float*)d_in[0];   // output: [64,1024,1024] f32
    const float* tgtp  = (const float*)d_in[1];   // target: [64,1024,1024] f32
    // d_in[2] = patch_size (16, assumed by the tiling), d_in[3] = filter_rate
    const int*   frate = (const int*)d_in[3];

    float* partials = (float*)d_ws;               // 4096 * 2 floats = 32 KB

    fpl_stage1<<<kBlocks, 256, 0, stream>>>(outp, tgtp, frate, partials);
    fpl_stage2<<<1, 256, 0, stream>>>(partials, (float*)d_out);
}